// SampledKWinners2d_88012469830609
// MI455X (gfx1250) — compile-verified
//
#include <hip/hip_runtime.h>
#include <hip/hip_bf16.h>

// ---------------------------------------------------------------------------
// SampledKWinners2d: per-row (B=64) top-k (k~52429) of score = x/10 + gumbel
// over n = 128*64*64 = 524288 elements, then out = relu(x) * mask.
// Strategy: MSB-first 8-bit radix select on order-preserving u32 keys.
// Keys live in d_out (128 MiB -> resident in MI455X's 192 MB L2), x/gumbel
// streamed with non-temporal loads. All select logic on-device (graph-safe).
// ---------------------------------------------------------------------------

typedef float    v4f __attribute__((ext_vector_type(4)));
typedef unsigned v4u __attribute__((ext_vector_type(4)));
typedef int      v4i __attribute__((ext_vector_type(4)));

typedef __attribute__((address_space(1))) v4i gv4i;   // global int4
typedef __attribute__((address_space(3))) v4i lv4i;   // LDS int4

static constexpr int B              = 64;
static constexpr int N              = 128 * 64 * 64;   // 524288 per row
static constexpr int NV             = N / 4;           // float4 per row
static constexpr int VEC_PER_BLOCK  = 1024;            // float4 per block
static constexpr int BLOCKS_PER_ROW = NV / VEC_PER_BLOCK; // 128
static constexpr int THREADS        = 256;             // 8 waves (wave32)

// workspace layout (uint32 words)
static constexpr int HIST_WORDS = B * 256;  // per-row 256-bin histograms
static constexpr int PREFIX_OFF = HIST_WORDS;        // [B] current key prefix
static constexpr int KK_OFF     = HIST_WORDS + B;    // [B] remaining k
static constexpr int THR_OFF    = HIST_WORDS + 2*B;  // [B] final threshold key

#if defined(__has_builtin)
#if __has_builtin(__builtin_amdgcn_global_load_async_to_lds_b128) && \
    __has_builtin(__builtin_amdgcn_s_wait_asynccnt)
#define USE_ASYNC_LDS 1
#endif
#endif

__device__ __forceinline__ unsigned score_key(float x, float g) {
    float s = x / 10.0f + g;                    // matches reference logits+gumbel
    unsigned u = __float_as_uint(s);
    // order-preserving map: larger float -> larger unsigned
    return (u & 0x80000000u) ? ~u : (u | 0x80000000u);
}

// ---------------------------------------------------------------------------
// init: zero histograms, set per-row prefix=0, kk=k
// ---------------------------------------------------------------------------
__global__ void init_kernel(unsigned* __restrict__ ws, const int* __restrict__ kptr) {
    const int row = blockIdx.x, tid = threadIdx.x;
    ws[row * 256 + tid] = 0u;
    if (row == 0 && tid < B) {
        ws[PREFIX_OFF + tid] = 0u;
        ws[KK_OFF + tid]     = (unsigned)(*kptr);
        ws[THR_OFF + tid]    = 0u;
    }
}

// ---------------------------------------------------------------------------
// pass 0: compute keys, store to d_out (stays in L2), histogram of byte 3
// ---------------------------------------------------------------------------
__global__ void pass0_kernel(const float* __restrict__ x, const float* __restrict__ g,
                             unsigned* __restrict__ keys, unsigned* __restrict__ ws) {
    __shared__ unsigned lh[256];
    const int tid = threadIdx.x;
    const int row = blockIdx.y;
    lh[tid] = 0u;
    __syncthreads();

    const long base = (long)row * NV + (long)blockIdx.x * VEC_PER_BLOCK;
    const v4f* xp = (const v4f*)x + base;
    const v4f* gp = (const v4f*)g + base;
    v4u*       kp = (v4u*)keys + base;

    for (int i = tid; i < VEC_PER_BLOCK; i += THREADS) {
        __builtin_prefetch(xp + i + THREADS, 0, 0);   // -> global_prefetch_b8
        __builtin_prefetch(gp + i + THREADS, 0, 0);
        v4f xv = __builtin_nontemporal_load(xp + i);  // dead after this pass
        v4f gv = __builtin_nontemporal_load(gp + i);
        v4u kv;
#pragma unroll
        for (int j = 0; j < 4; ++j) {
            unsigned u = score_key(xv[j], gv[j]);
            kv[j] = u;
            atomicAdd(&lh[u >> 24], 1u);
        }
        kp[i] = kv;                                   // temporal: keep in L2
    }
    __syncthreads();
    unsigned c = lh[tid];
    if (c) atomicAdd(&ws[row * 256 + tid], c);
}

// ---------------------------------------------------------------------------
// refinement histogram for byte SHIFT/8 among keys matching current prefix
// ---------------------------------------------------------------------------
template <int SHIFT>
__global__ void histo_kernel(const unsigned* __restrict__ keys, unsigned* __restrict__ ws) {
    __shared__ unsigned lh[256];
    const int tid = threadIdx.x;
    const int row = blockIdx.y;
    const unsigned pfx = ws[PREFIX_OFF + row];
    lh[tid] = 0u;
    __syncthreads();

    const long base = (long)row * NV + (long)blockIdx.x * VEC_PER_BLOCK;
    const v4u* kp = (const v4u*)keys + base;

#if defined(USE_ASYNC_LDS)
    // Stage this block's 16 KiB key tile into LDS with async-DMA loads
    // (GLOBAL_LOAD_ASYNC_TO_LDS_B128, tracked by ASYNCcnt).
    __shared__ __align__(16) unsigned stage[VEC_PER_BLOCK * 4];
#pragma unroll
    for (int i = 0; i < VEC_PER_BLOCK / THREADS; ++i) {
        const int slot = i * THREADS + tid;
        __builtin_amdgcn_global_load_async_to_lds_b128(
            (gv4i*)(kp + slot), (lv4i*)&stage[slot * 4], 0, 0);
    }
    __builtin_amdgcn_s_wait_asynccnt(0);
#pragma unroll
    for (int i = 0; i < VEC_PER_BLOCK / THREADS; ++i) {
        const int slot = i * THREADS + tid;
        v4u kv = *(const v4u*)&stage[slot * 4];
#pragma unroll
        for (int j = 0; j < 4; ++j) {
            unsigned u = kv[j];
            if (((u ^ pfx) >> (SHIFT + 8)) == 0u)
                atomicAdd(&lh[(u >> SHIFT) & 0xFFu], 1u);
        }
    }
#else
    for (int i = tid; i < VEC_PER_BLOCK; i += THREADS) {
        v4u kv = kp[i];                               // L2-resident
#pragma unroll
        for (int j = 0; j < 4; ++j) {
            unsigned u = kv[j];
            if (((u ^ pfx) >> (SHIFT + 8)) == 0u)
                atomicAdd(&lh[(u >> SHIFT) & 0xFFu], 1u);
        }
    }
#endif
    __syncthreads();
    unsigned c = lh[tid];
    if (c) atomicAdd(&ws[row * 256 + tid], c);
}

// ---------------------------------------------------------------------------
// per-row digit selection: suffix-scan 256 bins, pick digit holding k-th
// largest, update prefix/kk, zero histogram for the next pass.
// ---------------------------------------------------------------------------
template <int SHIFT>
__global__ void select_kernel(unsigned* __restrict__ ws) {
    __shared__ unsigned s[256];
    const int row = blockIdx.x, tid = threadIdx.x;
    unsigned* hist = ws + row * 256;
    unsigned c = hist[tid];
    hist[tid] = 0u;                 // reset for next histogram pass
    s[tid] = c;
    __syncthreads();
    // Hillis-Steele suffix sums: s[t] = sum_{j>=t} count[j]
    for (int off = 1; off < 256; off <<= 1) {
        unsigned v = (tid + off < 256) ? s[tid + off] : 0u;
        __syncthreads();
        s[tid] += v;
        __syncthreads();
    }
    const unsigned kk = ws[KK_OFF + row];
    const unsigned Sm = s[tid];
    const unsigned Sn = (tid < 255) ? s[tid + 1] : 0u;
    __syncthreads();
    if (Sm >= kk && Sn < kk) {      // exactly one thread (S non-increasing)
        unsigned np = ws[PREFIX_OFF + row] | ((unsigned)tid << SHIFT);
        ws[PREFIX_OFF + row] = np;
        ws[KK_OFF + row]     = kk - Sn;
        if (SHIFT == 0) ws[THR_OFF + row] = np;  // final 32-bit threshold key
    }
}

// ---------------------------------------------------------------------------
// apply: out = (key >= thresh) ? relu(x) : 0 ; overwrites keys in d_out
// ---------------------------------------------------------------------------
__global__ void apply_kernel(const float* __restrict__ x, unsigned* out,
                             const unsigned* __restrict__ ws) {
    const int tid = threadIdx.x;
    const int row = blockIdx.y;
    const unsigned thr = ws[THR_OFF + row];

    const long base = (long)row * NV + (long)blockIdx.x * VEC_PER_BLOCK;
    const v4f* xp = (const v4f*)x + base;
    v4u*       ob = (v4u*)out + base;    // holds keys; overwritten with output

    for (int i = tid; i < VEC_PER_BLOCK; i += THREADS) {
        v4u kv = ob[i];                                    // L2-resident key
        v4f xv = __builtin_nontemporal_load(xp + i);
        v4f o;
#pragma unroll
        for (int j = 0; j < 4; ++j)
            o[j] = (kv[j] >= thr) ? fmaxf(xv[j], 0.0f) : 0.0f;
        __builtin_nontemporal_store(o, (v4f*)(ob + i));    // streaming store
    }
}

// ---------------------------------------------------------------------------
extern "C" void kernel_launch(void* const* d_in, const int* in_sizes, int n_in,
                              void* d_out, int out_size, void* d_ws, size_t ws_size,
                              hipStream_t stream) {
    (void)in_sizes; (void)n_in; (void)out_size; (void)ws_size;
    const float* x = (const float*)d_in[0];
    const float* g = (const float*)d_in[1];
    const int*   k = (const int*)d_in[2];
    unsigned* keys = (unsigned*)d_out;   // keys, later the float output
    unsigned* ws   = (unsigned*)d_ws;

    const dim3 big(BLOCKS_PER_ROW, B);

    init_kernel<<<B, THREADS, 0, stream>>>(ws, k);
    pass0_kernel<<<big, THREADS, 0, stream>>>(x, g, keys, ws);
    select_kernel<24><<<B, THREADS, 0, stream>>>(ws);
    histo_kernel<16><<<big, THREADS, 0, stream>>>(keys, ws);
    select_kernel<16><<<B, THREADS, 0, stream>>>(ws);
    histo_kernel<8><<<big, THREADS, 0, stream>>>(keys, ws);
    select_kernel<8><<<B, THREADS, 0, stream>>>(ws);
    histo_kernel<0><<<big, THREADS, 0, stream>>>(keys, ws);
    select_kernel<0><<<B, THREADS, 0, stream>>>(ws);
    apply_kernel<<<big, THREADS, 0, stream>>>(x, keys, ws);
}